// _SelectiveSSM_58583353917470
// MI455X (gfx1250) — compile-verified
//
#include <hip/hip_runtime.h>
#include <cstdint>

#define D_MODEL 1024
#define D_STATE 16
#define DT_RANK 64
#define B_SZ 2
#define T_LEN 2048
#define BT (B_SZ * T_LEN)                 // 4096 rows
#define P_DIM (DT_RANK + 2 * D_STATE)     // 96
#define NCHUNK 16
#define CHUNK (T_LEN / NCHUNK)            // 128

typedef float v2f __attribute__((ext_vector_type(2)));
typedef float v8f __attribute__((ext_vector_type(8)));

// ---------------------------------------------------------------------------
// Kernel 1: xz[m, p] = sum_k x[m,k] * Wx[p,k]     (M=4096, N=96, K=1024)
// One wave32 per 16x16 tile, V_WMMA_F32_16X16X4_F32, K-step 4.
// A layout: lanes 0-15 hold M=lane, K={0,1}; lanes 16-31 hold M=lane-16, K={2,3}
// B layout (KxN, mirrored): lanes 0-15 hold N=lane, K={0,1}; lanes 16-31 K={2,3}
// D layout: VGPR r holds (M = r + 8*(lane>>4), N = lane&15)
// ---------------------------------------------------------------------------
__global__ __launch_bounds__(128) void k_xproj_wmma(
    const float* __restrict__ x, const float* __restrict__ Wx,
    float* __restrict__ xz) {
  const int lane = threadIdx.x & 31;
  const int wave = threadIdx.x >> 5;
  const int tile = blockIdx.x * 4 + wave;
  const int tilesM = BT / 16;                       // 256
  const int tm = tile % tilesM;
  const int tn = tile / tilesM;                     // 0..5
  if (tn >= P_DIM / 16) return;                     // wave-uniform guard
  const int m0 = tm * 16, n0 = tn * 16;
  const int half = lane >> 4;
  const int l15  = lane & 15;
  const int kk   = half * 2;

  v8f acc = {};
  const float* aRow = x  + (size_t)(m0 + l15) * D_MODEL + kk;
  const float* bRow = Wx + (size_t)(n0 + l15) * D_MODEL + kk;
#pragma unroll 8
  for (int k0 = 0; k0 < D_MODEL; k0 += 4) {
    v2f a, b;
    a.x = aRow[k0]; a.y = aRow[k0 + 1];
    b.x = bRow[k0]; b.y = bRow[k0 + 1];
    acc = __builtin_amdgcn_wmma_f32_16x16x4_f32(
        false, a, false, b, (short)0, acc, false, false);
  }
  const int rOff = half * 8;
#pragma unroll
  for (int r = 0; r < 8; ++r)
    xz[(size_t)(m0 + rOff + r) * P_DIM + n0 + l15] = acc[r];
}

// ---------------------------------------------------------------------------
// Kernel 2: dt[m, n] = softplus( sum_r xz[m, r] * Wdt[n, r] + bdt[n] )
//           (M=4096, N=1024, K=64)
// ---------------------------------------------------------------------------
__global__ __launch_bounds__(128) void k_dt_wmma(
    const float* __restrict__ xz, const float* __restrict__ Wdt,
    const float* __restrict__ bdt, float* __restrict__ dt) {
  const int lane = threadIdx.x & 31;
  const int wave = threadIdx.x >> 5;
  const int tile = blockIdx.x * 4 + wave;
  const int tilesM = BT / 16;                       // 256
  const int tm = tile % tilesM;
  const int tn = tile / tilesM;                     // 0..63
  if (tn >= D_MODEL / 16) return;
  const int m0 = tm * 16, n0 = tn * 16;
  const int half = lane >> 4;
  const int l15  = lane & 15;
  const int kk   = half * 2;

  v8f acc = {};
  const float* aRow = xz  + (size_t)(m0 + l15) * P_DIM   + kk;
  const float* bRow = Wdt + (size_t)(n0 + l15) * DT_RANK + kk;
#pragma unroll
  for (int k0 = 0; k0 < DT_RANK; k0 += 4) {
    v2f a, b;
    a.x = aRow[k0]; a.y = aRow[k0 + 1];
    b.x = bRow[k0]; b.y = bRow[k0 + 1];
    acc = __builtin_amdgcn_wmma_f32_16x16x4_f32(
        false, a, false, b, (short)0, acc, false, false);
  }
  const int rOff = half * 8;
#pragma unroll
  for (int r = 0; r < 8; ++r) {
    const int n = n0 + l15;
    const float v = acc[r] + bdt[n];
    const float sp = (v > 20.f) ? v : __logf(1.f + __expf(v));
    dt[(size_t)(m0 + rOff + r) * D_MODEL + n] = sp;
  }
}

// ---------------------------------------------------------------------------
// Scan pass A: per (b, d, s, chunk) compute chunk-local partial state Hloc
// and chunk decay product Pw = exp(A_s * sum_t dt) (algebraic product fold).
// Lane layout: s fastest (16 lanes), 2 channels per wave.
// ---------------------------------------------------------------------------
__global__ __launch_bounds__(256) void k_scanA(
    const float* __restrict__ x, const float* __restrict__ xz,
    const float* __restrict__ dt, const float* __restrict__ A_log,
    float* __restrict__ Pw, float* __restrict__ Hloc) {
  const int gid = blockIdx.x * 256 + threadIdx.x;   // 0 .. 2^19-1
  const int s = gid & 15;
  const int d = (gid >> 4) & (D_MODEL - 1);
  const int c = (gid >> 14) & (NCHUNK - 1);
  const int b = gid >> 18;

  const float As = -__expf(A_log[d * D_STATE + s]);
  const int t0 = c * CHUNK;
  const size_t rowD = (size_t)(b * T_LEN + t0) * D_MODEL + d;
  const size_t rowP = (size_t)(b * T_LEN + t0) * P_DIM;

  float h = 0.f, dtsum = 0.f;
  for (int t = 0; t < CHUNK; ++t) {
    __builtin_prefetch(dt + rowD + (size_t)(t + 16) * D_MODEL, 0, 3);
    const float dtv = dt[rowD + (size_t)t * D_MODEL];
    const float xv  = x [rowD + (size_t)t * D_MODEL];
    const float Bv  = xz[rowP + (size_t)t * P_DIM + DT_RANK + s];
    const float a   = __expf(dtv * As);
    h = __builtin_fmaf(a, h, dtv * Bv * xv);
    dtsum += dtv;
  }
  const size_t widx = (((size_t)b * NCHUNK + c) * D_MODEL + d) * D_STATE + s;
  Pw[widx]   = __expf(As * dtsum);
  Hloc[widx] = h;
}

// ---------------------------------------------------------------------------
// Scan pass B: sequential combine of the 16 chunk partials per (b, d, s).
// Hinit[c] = state entering chunk c.
// ---------------------------------------------------------------------------
__global__ __launch_bounds__(256) void k_scanB(
    const float* __restrict__ Pw, const float* __restrict__ Hloc,
    float* __restrict__ Hinit) {
  const int gid = blockIdx.x * 256 + threadIdx.x;   // 0 .. 32767
  const int s = gid & 15;
  const int d = (gid >> 4) & (D_MODEL - 1);
  const int b = gid >> 14;
  float h = 0.f;
#pragma unroll
  for (int c = 0; c < NCHUNK; ++c) {
    const size_t widx = (((size_t)b * NCHUNK + c) * D_MODEL + d) * D_STATE + s;
    Hinit[widx] = h;
    h = __builtin_fmaf(Pw[widx], h, Hloc[widx]);
  }
}

// ---------------------------------------------------------------------------
// Scan pass C: re-run each chunk from its exact initial state; reduce over s
// (16 contiguous lanes) via shfl_xor butterfly; fuse out = y + Dp * x.
// ---------------------------------------------------------------------------
__global__ __launch_bounds__(256) void k_scanC(
    const float* __restrict__ x, const float* __restrict__ xz,
    const float* __restrict__ dt, const float* __restrict__ A_log,
    const float* __restrict__ Dp, const float* __restrict__ Hinit,
    float* __restrict__ out) {
  const int gid = blockIdx.x * 256 + threadIdx.x;
  const int s = gid & 15;
  const int d = (gid >> 4) & (D_MODEL - 1);
  const int c = (gid >> 14) & (NCHUNK - 1);
  const int b = gid >> 18;

  const float As  = -__expf(A_log[d * D_STATE + s]);
  const float Dpd = Dp[d];
  const int t0 = c * CHUNK;
  const size_t rowD = (size_t)(b * T_LEN + t0) * D_MODEL + d;
  const size_t rowP = (size_t)(b * T_LEN + t0) * P_DIM;
  const size_t widx = (((size_t)b * NCHUNK + c) * D_MODEL + d) * D_STATE + s;

  float h = Hinit[widx];
  for (int t = 0; t < CHUNK; ++t) {
    __builtin_prefetch(x + rowD + (size_t)(t + 16) * D_MODEL, 0, 3);
    const float dtv = dt[rowD + (size_t)t * D_MODEL];
    const float xv  = x [rowD + (size_t)t * D_MODEL];
    const float Bv  = xz[rowP + (size_t)t * P_DIM + DT_RANK + s];
    const float Cv  = xz[rowP + (size_t)t * P_DIM + DT_RANK + D_STATE + s];
    const float a   = __expf(dtv * As);
    h = __builtin_fmaf(a, h, dtv * Bv * xv);
    float yc = h * Cv;
    yc += __shfl_xor(yc, 1, 32);
    yc += __shfl_xor(yc, 2, 32);
    yc += __shfl_xor(yc, 4, 32);
    yc += __shfl_xor(yc, 8, 32);
    if (s == 0)
      out[rowD + (size_t)t * D_MODEL] = yc + Dpd * xv;
  }
}

// ---------------------------------------------------------------------------
extern "C" void kernel_launch(void* const* d_in, const int* in_sizes, int n_in,
                              void* d_out, int out_size, void* d_ws, size_t ws_size,
                              hipStream_t stream) {
  (void)in_sizes; (void)n_in; (void)out_size; (void)ws_size;
  const float* x    = (const float*)d_in[0];
  const float* Wx   = (const float*)d_in[1];
  const float* Wdt  = (const float*)d_in[2];
  const float* bdt  = (const float*)d_in[3];
  const float* Alog = (const float*)d_in[4];
  const float* Dp   = (const float*)d_in[5];
  float* out = (float*)d_out;

  float* ws    = (float*)d_ws;
  float* xz    = ws;                                          // 4096*96
  float* dt    = xz + (size_t)BT * P_DIM;                     // 4096*1024
  float* Pw    = dt + (size_t)BT * D_MODEL;                   // 2*16*1024*16
  const size_t nScan = (size_t)B_SZ * NCHUNK * D_MODEL * D_STATE; // 524288
  float* Hloc  = Pw + nScan;
  float* Hinit = Hloc + nScan;

  k_xproj_wmma<<<(BT / 16) * (P_DIM / 16) / 4, 128, 0, stream>>>(x, Wx, xz);
  k_dt_wmma<<<(BT / 16) * (D_MODEL / 16) / 4, 128, 0, stream>>>(xz, Wdt, bdt, dt);
  k_scanA<<<(int)(nScan / 256), 256, 0, stream>>>(x, xz, dt, Alog, Pw, Hloc);
  k_scanB<<<(B_SZ * D_MODEL * D_STATE) / 256, 256, 0, stream>>>(Pw, Hloc, Hinit);
  k_scanC<<<(int)(nScan / 256), 256, 0, stream>>>(x, xz, dt, Alog, Dp, Hinit, out);
}